// FilterDetections_58102317580692
// MI455X (gfx1250) — compile-verified
//
#include <hip/hip_runtime.h>
#include <stdint.h>

// ---------------- problem constants (match reference) ----------------
#define NEG_SENT  (-1.0e9f)
#define SCORE_TH  (0.05f)
#define IOU_TH    (0.5f)
#define MAXDET    100
#define NBOX      10000
#define NCLS      80
#define NIMG      2
#define NTHREADS  1024
#define NWAVES    (NTHREADS / 32)
#define KELEM     ((NBOX + NTHREADS - 1) / NTHREADS)      // 10 boxes per thread
#define NCAND     (NCLS * MAXDET)                         // 8000
#define KCAND     ((NCAND + NTHREADS - 1) / NTHREADS)     // 8 candidates per thread

typedef unsigned int u32;
typedef u32 __attribute__((ext_vector_type(4))) u32x4;
typedef int  __attribute__((ext_vector_type(4))) i32x4;
typedef int  __attribute__((ext_vector_type(8))) i32x8;

#if defined(__HIP_DEVICE_COMPILE__) && defined(__has_builtin)
#if __has_builtin(__builtin_amdgcn_tensor_load_to_lds) && __has_builtin(__builtin_amdgcn_s_wait_tensorcnt)
#define USE_TDM 1
#endif
#endif
#ifndef USE_TDM
#define USE_TDM 0
#endif

#if USE_TDM
// Issue one TDM 2D tensor-load descriptor: global (row-major, 4-byte elems,
// row stride `stride0` elems) -> LDS at byte offset lds_off. Tile is
// tile_d0 x tile_d1 elements. Packing per CDNA5 ISA D# groups 0/1.
__device__ __forceinline__ void tdm_load_2d(u32 lds_off, const void* gptr,
                                            u32 tensor_d0, u32 tensor_d1,
                                            u32 tile_d0,   u32 tile_d1,
                                            u32 stride0)
{
    uint64_t ga = (uint64_t)(uintptr_t)gptr;
    u32x4 g0;
    g0.x = 1u;                                   // count=1, user descriptor
    g0.y = lds_off;                              // lds_addr (bytes)
    g0.z = (u32)(ga & 0xffffffffu);              // global_addr[31:0]
    g0.w = (u32)((ga >> 32) & 0x01ffffffu)       // global_addr[56:32]
         | (2u << 30);                           // type = 2 ("image")
    i32x8 g1;
    g1[0] = (int)(2u << 16);                     // data_size = 4B; wg_mask=0
    g1[1] = (int)((tensor_d0 & 0xffffu) << 16);  // tensor_dim0[15:0] -> bits 63:48
    g1[2] = (int)((tensor_d0 >> 16) | ((tensor_d1 & 0xffffu) << 16));
    g1[3] = (int)((tensor_d1 >> 16) | ((tile_d0 & 0xffffu) << 16)); // tile_dim0
    g1[4] = (int)(tile_d1 & 0xffffu);            // tile_dim1 (tile_dim2 = 0)
    g1[5] = (int)stride0;                        // tensor_dim0_stride[31:0]
    g1[6] = 0;                                   // stride0 hi / stride1 lo
    g1[7] = 0;                                   // stride1 hi
    i32x4 z4 = {0, 0, 0, 0};
#if __clang_major__ >= 23
    i32x8 z8 = {0, 0, 0, 0, 0, 0, 0, 0};
    __builtin_amdgcn_tensor_load_to_lds(g0, g1, z4, z4, z8, 0);
#else
    __builtin_amdgcn_tensor_load_to_lds(g0, g1, z4, z4, 0);
#endif
}
#endif // USE_TDM

// argmax (score, lowest index wins ties) across a 32-lane wave
__device__ __forceinline__ void wave_argmax(float& s, int& i)
{
    #pragma unroll
    for (int off = 16; off > 0; off >>= 1) {
        float so = __shfl_down(s, (unsigned)off, 32);
        int   io = __shfl_down(i, (unsigned)off, 32);
        if (so > s || (so == s && io < i)) { s = so; i = io; }
    }
}

// ---------------------------------------------------------------------
// Kernel A: one workgroup per (image, class). TDM lands boxes (160 KB,
// contiguous) and per-class scores (stride-320B gather -> packed 40 KB)
// in LDS; each thread then owns a 10-element slice in REGISTERS
// (score + box + area). 100 greedy NMS steps run entirely out of VGPRs;
// LDS is touched only for the 32-entry cross-wave argmax and the
// winner-box gather. 2 barriers per step.
// ---------------------------------------------------------------------
__global__ void __launch_bounds__(NTHREADS)
nms_class_kernel(const float* __restrict__ boxes,   // (B, N, 4)
                 const float* __restrict__ cls,     // (B, N, C)
                 float* __restrict__ keep_scores,   // (B*C, 100)
                 float* __restrict__ keep_boxes)    // (B*C, 100, 4)
{
    const int bc = blockIdx.x;       // b*C + c
    const int b  = bc / NCLS;
    const int c  = bc % NCLS;
    const int t  = threadIdx.x;

    extern __shared__ float smem[];
    float* bx = smem;                // 4*NBOX floats, AoS (x1,y1,x2,y2)
    float* sc = smem + 4 * NBOX;     // NBOX floats

    __shared__ float  wred_s[NWAVES];
    __shared__ int    wred_i[NWAVES];
    __shared__ float4 bshare;

#if USE_TDM
    if (t < 32) {   // wave 0 issues both DMA descriptors (EXEC ignored by TDM)
        u32 lds_base = (u32)(uintptr_t)(void*)smem;   // LDS flat addr low bits = group offset
        // contiguous 160 KB box tile
        tdm_load_2d(lds_base,
                    boxes + (size_t)b * 4 * NBOX,
                    4 * NBOX, 1, 4 * NBOX, 1, 4 * NBOX);
        // strided score gather: one elem per row, 10000 rows, row stride C
        tdm_load_2d(lds_base + 4u * 4u * NBOX,
                    cls + (size_t)b * NBOX * NCLS + c,
                    NCLS, NBOX, 1, NBOX, NCLS);
        __builtin_amdgcn_s_wait_tensorcnt(0);
    }
    __syncthreads();
#else
    for (int i = t; i < 4 * NBOX; i += NTHREADS)
        bx[i] = boxes[(size_t)b * 4 * NBOX + i];
    for (int i = t; i < NBOX; i += NTHREADS)
        sc[i] = cls[((size_t)b * NBOX + i) * NCLS + c];
    __syncthreads();
#endif

    // ---- pull this thread's slice into registers (threshold fused) ----
    const float4* bx4 = (const float4*)bx;
    float  sreg[KELEM];
    float4 breg[KELEM];
    float  areg[KELEM];
    #pragma unroll
    for (int k = 0; k < KELEM; ++k) {
        const int i = t + k * NTHREADS;
        if (i < NBOX) {
            float s  = sc[i];
            sreg[k]  = (s > SCORE_TH) ? s : NEG_SENT;
            float4 q = bx4[i];
            breg[k]  = q;
            areg[k]  = (q.z - q.x) * (q.w - q.y);
        } else {
            sreg[k] = -3.0e38f;
            breg[k] = make_float4(0.0f, 0.0f, 0.0f, 0.0f);
            areg[k] = 0.0f;
        }
    }

    const int lane = t & 31;
    const int wave = t >> 5;

    for (int iter = 0; iter < MAXDET; ++iter) {
        // --- register argmax (lowest-index tie-break) ---
        float bs = -3.0e38f;
        int   bi = 0x7fffffff;
        #pragma unroll
        for (int k = 0; k < KELEM; ++k) {
            if (sreg[k] > bs) { bs = sreg[k]; bi = t + k * NTHREADS; }
        }
        wave_argmax(bs, bi);
        if (lane == 0) { wred_s[wave] = bs; wred_i[wave] = bi; }
        __syncthreads();
        if (wave == 0) {
            bs = wred_s[lane];
            bi = wred_i[lane];
            wave_argmax(bs, bi);
            if (lane == 0) {
                float4 bb = bx4[bi];         // gather winner box from LDS
                bshare = bb;
                keep_scores[(size_t)bc * MAXDET + iter] = bs;
                float* kb = keep_boxes + ((size_t)bc * MAXDET + iter) * 4;
                kb[0] = bb.x; kb[1] = bb.y; kb[2] = bb.z; kb[3] = bb.w;
            }
        }
        __syncthreads();

        // --- suppress in registers: IoU(best, mine) > 0.5 -> NEG ---
        const float4 Bb = bshare;
        const float  a1 = (Bb.z - Bb.x) * (Bb.w - Bb.y);
        #pragma unroll
        for (int k = 0; k < KELEM; ++k) {
            const float4 q = breg[k];
            float ix1 = fmaxf(Bb.x, q.x), iy1 = fmaxf(Bb.y, q.y);
            float ix2 = fminf(Bb.z, q.z), iy2 = fminf(Bb.w, q.w);
            float inter = fmaxf(ix2 - ix1, 0.0f) * fmaxf(iy2 - iy1, 0.0f);
            float uni   = fmaxf(a1 + areg[k] - inter, 1e-8f);
            // iou > 0.5  <=>  inter > 0.5*uni   (uni > 0)
            if (inter > IOU_TH * uni) sreg[k] = NEG_SENT;
        }
    }
}

// ---------------------------------------------------------------------
// Kernel B: one workgroup per image. top-100 of 8000 candidates via
// repeated register argmax (8 candidates/thread; matches jax top_k
// lowest-index tie ordering). Owner thread retires the picked entry.
// ---------------------------------------------------------------------
__global__ void __launch_bounds__(NTHREADS)
topk_kernel(const float* __restrict__ keep_scores,  // (B, C*100)
            const float* __restrict__ keep_boxes,   // (B, C*100, 4)
            float* __restrict__ out)                // [boxes 800 | scores 200 | labels 200]
{
    const int b = blockIdx.x;
    const int t = threadIdx.x;

    __shared__ float wred_s[NWAVES];
    __shared__ int   wred_i[NWAVES];
    __shared__ int   pick_sh;
    __shared__ float ps_sh;

    float sreg[KCAND];
    #pragma unroll
    for (int k = 0; k < KCAND; ++k) {
        const int i = t + k * NTHREADS;
        sreg[k] = (i < NCAND) ? keep_scores[(size_t)b * NCAND + i] : -3.0e38f;
    }

    float* ob = out;                          // (B,100,4)
    float* os = out + NIMG * MAXDET * 4;      // (B,100)
    float* ol = os + NIMG * MAXDET;           // (B,100) labels as float

    const int lane = t & 31;
    const int wave = t >> 5;

    for (int m = 0; m < MAXDET; ++m) {
        float bs = -3.0e38f;
        int   bi = 0x7fffffff;
        #pragma unroll
        for (int k = 0; k < KCAND; ++k) {
            if (sreg[k] > bs) { bs = sreg[k]; bi = t + k * NTHREADS; }
        }
        wave_argmax(bs, bi);
        if (lane == 0) { wred_s[wave] = bs; wred_i[wave] = bi; }
        __syncthreads();
        if (wave == 0) {
            bs = wred_s[lane];
            bi = wred_i[lane];
            wave_argmax(bs, bi);
            if (lane == 0) { pick_sh = bi; ps_sh = bs; }
        }
        __syncthreads();

        const int pick = pick_sh;
        // owner thread retires the picked candidate from its registers
        if ((pick & (NTHREADS - 1)) == t) sreg[pick >> 10] = -3.0e9f;

        if (t == 0) {
            const float s      = ps_sh;
            const bool  valid  = (s > SCORE_TH);
            const int   cls_id = pick / MAXDET;
            const float* kb = keep_boxes + ((size_t)b * NCAND + pick) * 4;
            float* dob = ob + ((size_t)b * MAXDET + m) * 4;
            dob[0] = valid ? kb[0] : -1.0f;
            dob[1] = valid ? kb[1] : -1.0f;
            dob[2] = valid ? kb[2] : -1.0f;
            dob[3] = valid ? kb[3] : -1.0f;
            os[b * MAXDET + m] = valid ? s : -1.0f;
            ol[b * MAXDET + m] = valid ? (float)cls_id : -1.0f;
        }
        // wred reuse is protected by the two barriers above (writes to wred
        // for iteration m+1 happen only after all threads passed barrier 2)
    }
}

extern "C" void kernel_launch(void* const* d_in, const int* in_sizes, int n_in,
                              void* d_out, int out_size, void* d_ws, size_t ws_size,
                              hipStream_t stream)
{
    const float* boxes = (const float*)d_in[0];   // (B, N, 4) f32
    const float* cls   = (const float*)d_in[1];   // (B, N, C) f32

    float* ws          = (float*)d_ws;
    float* keep_scores = ws;                                // B*C*100   = 16000 f32
    float* keep_boxes  = ws + (size_t)NIMG * NCLS * MAXDET; // B*C*100*4 = 64000 f32

    const size_t dyn_lds = (size_t)(4 * NBOX + NBOX) * sizeof(float); // 200000 B

    nms_class_kernel<<<NIMG * NCLS, NTHREADS, dyn_lds, stream>>>(
        boxes, cls, keep_scores, keep_boxes);
    topk_kernel<<<NIMG, NTHREADS, 0, stream>>>(
        keep_scores, keep_boxes, (float*)d_out);
}